// InfoNCE_22832046146017
// MI455X (gfx1250) — compile-verified
//
#include <hip/hip_runtime.h>
#include <hip/hip_bf16.h>
#include <math.h>

typedef float v2f __attribute__((ext_vector_type(2)));
typedef float v8f __attribute__((ext_vector_type(8)));

#define N_SAMP 1024
#define XDIM   128
#define HID    256
#define TI     4      // i-rows per block in the pairwise kernel

// ---------------------------------------------------------------------------
// Kernel 1: HxT = (x @ W1[:128])^T  (stored [k][j], 256x1024)
//           Hyb =  y @ W1[128:] + b1 (row-major, 1024x256)
// One wave32 per 16x16 output tile, K=128 via 32x V_WMMA_F32_16X16X4_F32.
// f32 WMMA keeps full precision (GEMM work is tiny; accuracy matters for MI).
// A frag (16x4 f32): lane-half h, VGPR v -> K = k0 + 2h + v, M = lane&15.
// B frag (4x16 f32): VGPR v -> K = k0 + 2h + v, N = lane&15.
// C/D (16x16 f32): VGPR r -> M = r + 8h, N = lane&15.  For the transposed
// store, each lane's 8 values are 8 consecutive rows of one column -> two
// aligned float4 (b128) stores.
// ---------------------------------------------------------------------------
__global__ void __launch_bounds__(128) gemm_front_wmma(
    const float* __restrict__ X, const float* __restrict__ Y,
    const float* __restrict__ W1, const float* __restrict__ b1,
    float* __restrict__ HxT, float* __restrict__ Hyb)
{
    const int lane = threadIdx.x & 31;
    const int wave = threadIdx.x >> 5;
    const int tile = blockIdx.x * 4 + wave;   // 0..2047
    const int mat  = tile >> 10;              // 0 = HxT, 1 = Hyb
    const int t    = tile & 1023;             // 64 m-tiles x 16 n-tiles
    const int m0   = (t >> 4) << 4;
    const int n0   = (t & 15) << 4;
    const int half = lane >> 4;
    const int l16  = lane & 15;

    const float* __restrict__ A = mat ? Y : X;
    const float* __restrict__ W = mat ? (W1 + XDIM * HID) : W1;

    v8f c = {};
    const float* __restrict__ arow = A + (m0 + l16) * XDIM + 2 * half;
    #pragma unroll 4
    for (int k = 0; k < XDIM; k += 4) {
        v2f a, b;
        a.x = arow[k];
        a.y = arow[k + 1];
        const float* __restrict__ wrow = W + (size_t)(k + 2 * half) * HID + n0 + l16;
        b.x = wrow[0];
        b.y = wrow[HID];
        c = __builtin_amdgcn_wmma_f32_16x16x4_f32(false, a, false, b,
                                                  (short)0, c, false, false);
    }

    if (mat) {
        const float bias = b1[n0 + l16];
        #pragma unroll
        for (int r = 0; r < 8; ++r) {
            const int row = m0 + r + 8 * half;
            Hyb[row * HID + n0 + l16] = c[r] + bias;
        }
    } else {
        // HxT[col * N + row], 8 consecutive rows per lane -> 2x b128 stores
        float4 lo, hi;
        lo.x = c[0]; lo.y = c[1]; lo.z = c[2]; lo.w = c[3];
        hi.x = c[4]; hi.y = c[5]; hi.z = c[6]; hi.w = c[7];
        float4* dst = (float4*)(HxT + (size_t)(n0 + l16) * N_SAMP + m0 + 8 * half);
        dst[0] = lo;
        dst[1] = hi;
    }
}

__device__ __forceinline__ float softplusf(float z) {
    // numerically stable: max(z,0) + log1p(exp(-|z|))
    return fmaxf(z, 0.0f) + log1pf(expf(-fabsf(z)));
}

// ---------------------------------------------------------------------------
// Kernel 2: block handles TI=4 rows i vs all 1024 j.
// s_ij = sum_k relu(Hyb[i,k] + Hx[j,k]) * w2[k];  T1 = softplus(s + b2).
// HxT layout makes the k-loop loads fully coalesced (512 B per wave per k):
// lane owns j = 4*tid..4*tid+3 via one float4.  Hyb rows + w2 sit in LDS and
// are pure broadcast reads.  48 VALU ops per 16 B loaded -> compute-bound.
// Diagonal (j == i) doubles as T0[i].  Per-row logsumexp via block reduction.
// ---------------------------------------------------------------------------
__global__ void __launch_bounds__(256) pairwise_lse_kernel(
    const float* __restrict__ HxT, const float* __restrict__ Hyb,
    const float* __restrict__ W2, const float* __restrict__ b2,
    float* __restrict__ rowLse, float* __restrict__ T0)
{
    __shared__ float sHy[TI * HID];
    __shared__ float sWw[HID];
    __shared__ float red[256];

    const int i0  = blockIdx.x * TI;
    const int tid = threadIdx.x;
    #pragma unroll
    for (int r = 0; r < TI; ++r)
        sHy[r * HID + tid] = Hyb[(i0 + r) * HID + tid];
    sWw[tid] = W2[tid];
    __syncthreads();

    const float bb = b2[0];
    float acc[TI][4];
    #pragma unroll
    for (int r = 0; r < TI; ++r)
        acc[r][0] = acc[r][1] = acc[r][2] = acc[r][3] = 0.f;

    const float4* __restrict__ hx = ((const float4*)HxT) + tid;  // j = 4*tid..+3
    #pragma unroll 4
    for (int k = 0; k < HID; ++k) {
        const float4 h = hx[k * (N_SAMP / 4)];
        const float  w = sWw[k];
        #pragma unroll
        for (int r = 0; r < TI; ++r) {
            const float y = sHy[r * HID + k];
            acc[r][0] += fmaxf(h.x + y, 0.f) * w;
            acc[r][1] += fmaxf(h.y + y, 0.f) * w;
            acc[r][2] += fmaxf(h.z + y, 0.f) * w;
            acc[r][3] += fmaxf(h.w + y, 0.f) * w;
        }
    }

    #pragma unroll 1
    for (int r = 0; r < TI; ++r) {
        const int i = i0 + r;
        const float s0 = softplusf(acc[r][0] + bb);
        const float s1 = softplusf(acc[r][1] + bb);
        const float s2 = softplusf(acc[r][2] + bb);
        const float s3 = softplusf(acc[r][3] + bb);

        // Diagonal term: h1[i,i] == relu(hx_i + hy_i + b1), so T0[i] = T1[i,i].
        if (tid == (i >> 2)) {
            const float diag[4] = {s0, s1, s2, s3};
            T0[i] = diag[i & 3];
        }

        // logsumexp over the 1024 T1 values of row i
        red[tid] = fmaxf(fmaxf(s0, s1), fmaxf(s2, s3));
        __syncthreads();
        for (int s = 128; s > 0; s >>= 1) {
            if (tid < s) red[tid] = fmaxf(red[tid], red[tid + s]);
            __syncthreads();
        }
        const float M = red[0];
        __syncthreads();

        red[tid] = expf(s0 - M) + expf(s1 - M) + expf(s2 - M) + expf(s3 - M);
        __syncthreads();
        for (int s = 128; s > 0; s >>= 1) {
            if (tid < s) red[tid] += red[tid + s];
            __syncthreads();
        }
        if (tid == 0) rowLse[i] = M + logf(red[0]);
        __syncthreads();
    }
}

// ---------------------------------------------------------------------------
// Kernel 3: lower = mean(T0) - (mean(rowLse) - log(N))
// ---------------------------------------------------------------------------
__global__ void __launch_bounds__(256) final_reduce_kernel(
    const float* __restrict__ rowLse, const float* __restrict__ T0,
    float* __restrict__ out)
{
    __shared__ float rA[256];
    __shared__ float rB[256];
    const int tid = threadIdx.x;
    float a = 0.f, b = 0.f;
    for (int r = tid; r < N_SAMP; r += 256) { a += T0[r]; b += rowLse[r]; }
    rA[tid] = a; rB[tid] = b;
    __syncthreads();
    for (int s = 128; s > 0; s >>= 1) {
        if (tid < s) { rA[tid] += rA[tid + s]; rB[tid] += rB[tid + s]; }
        __syncthreads();
    }
    if (tid == 0) {
        const float invn = 1.0f / (float)N_SAMP;
        out[0] = rA[0] * invn - (rB[0] * invn - logf((float)N_SAMP));
    }
}

extern "C" void kernel_launch(void* const* d_in, const int* in_sizes, int n_in,
                              void* d_out, int out_size, void* d_ws, size_t ws_size,
                              hipStream_t stream) {
    const float* x  = (const float*)d_in[0];   // (1024,128)
    const float* y  = (const float*)d_in[1];   // (1024,128)
    const float* W1 = (const float*)d_in[2];   // (256,256)
    const float* b1 = (const float*)d_in[3];   // (256,)
    const float* W2 = (const float*)d_in[4];   // (256,1)
    const float* b2 = (const float*)d_in[5];   // (1,)
    float* out = (float*)d_out;

    float* HxT    = (float*)d_ws;              // 256*1024 f32 (transposed hx)
    float* Hyb    = HxT + HID * N_SAMP;        // 1024*256 f32
    float* rowLse = Hyb + N_SAMP * HID;        // 1024 f32
    float* T0     = rowLse + N_SAMP;           // 1024 f32

    // 2048 16x16 tiles, 4 waves (tiles) per 128-thread block.
    gemm_front_wmma<<<512, 128, 0, stream>>>(x, y, W1, b1, HxT, Hyb);
    pairwise_lse_kernel<<<N_SAMP / TI, 256, 0, stream>>>(HxT, Hyb, W2, b2, rowLse, T0);
    final_reduce_kernel<<<1, 256, 0, stream>>>(rowLse, T0, out);
}